// VladPooling_38182259261790
// MI455X (gfx1250) — compile-verified
//
#include <hip/hip_runtime.h>
#include <math.h>

typedef float v2f __attribute__((ext_vector_type(2)));
typedef float v8f __attribute__((ext_vector_type(8)));

#define BATCH 32
#define CIN   512
#define HW    4096      // 16*256
#define NCLUS 10
#define KPAD  16
#define KKEEP 8
#define KCHUNK (HW / 8) // 512 hw elements per wave

// ---------------------------------------------------------------------------
// ws layout (floats):
//   S    [32][16][4096]  softmax (padded to 16 clusters)   = 2,097,152
//   WX   [32][16][512]   wx accumulator                    =   262,144
//   ASUM [32][16]        per-cluster soft-assign sums      =       512
// ---------------------------------------------------------------------------

__global__ void vlad_zero_asum(float* __restrict__ asum) {
    int i = blockIdx.x * blockDim.x + threadIdx.x;
    if (i < BATCH * KPAD) asum[i] = 0.0f;
}

// One thread per (b, hw) pixel: softmax over the 10 cluster scores,
// write padded S, block-reduce per-cluster sums -> one atomicAdd per block.
__global__ __launch_bounds__(256) void vlad_softmax(const float* __restrict__ score,
                                                    float* __restrict__ S,
                                                    float* __restrict__ asum) {
    int gid = blockIdx.x * 256 + threadIdx.x;
    int b   = gid >> 12;        // / 4096
    int hw  = gid & (HW - 1);

    const float* sp = score + (size_t)b * NCLUS * HW + hw;
    float s[NCLUS];
    float mx = -1e30f;
#pragma unroll
    for (int k = 0; k < NCLUS; ++k) { s[k] = sp[(size_t)k * HW]; mx = fmaxf(mx, s[k]); }
    float sum = 0.0f;
#pragma unroll
    for (int k = 0; k < NCLUS; ++k) { s[k] = __expf(s[k] - mx); sum += s[k]; }
    float inv = 1.0f / sum;

    float* Sp = S + (size_t)b * KPAD * HW + hw;
#pragma unroll
    for (int k = 0; k < NCLUS; ++k) { s[k] *= inv; Sp[(size_t)k * HW] = s[k]; }
#pragma unroll
    for (int k = NCLUS; k < KPAD; ++k) Sp[(size_t)k * HW] = 0.0f;  // zero padded rows

    __shared__ float red[256];
    for (int k = 0; k < NCLUS; ++k) {
        red[threadIdx.x] = s[k];
        __syncthreads();
        for (int off = 128; off > 0; off >>= 1) {
            if (threadIdx.x < off) red[threadIdx.x] += red[threadIdx.x + off];
            __syncthreads();
        }
        if (threadIdx.x == 0) atomicAdd(&asum[b * KPAD + k], red[0]);
        __syncthreads();
    }
}

// Per-batch GEMM  WX[b] (16x512) = S[b] (16x4096) @ F[b]^T (4096x512)
// Block = 8 waves; block handles (b, 64-channel tile); each wave owns a
// disjoint 512-wide K chunk; cross-wave reduction through LDS.
__global__ __launch_bounds__(256) void vlad_wmma_gemm(const float* __restrict__ F,
                                                      const float* __restrict__ S,
                                                      float* __restrict__ WX) {
    const int b     = blockIdx.x >> 3;
    const int ntile = blockIdx.x & 7;
    const int c0    = ntile * 64;

    const int wave  = threadIdx.x >> 5;   // 0..7 -> K chunk
    const int lane  = threadIdx.x & 31;
    const int kbase = wave * KCHUNK;
    const int mn    = lane & 15;          // M for A-frag, N for B-frag
    const int kk    = (lane >> 4) * 2;    // K sub-offset within the x4 step

    const float* Sp = S + ((size_t)b * KPAD + mn) * HW + kbase + kk;
    const float* Fp = F + ((size_t)b * CIN + c0 + mn) * HW + kbase + kk;

    v8f acc0 = {}; v8f acc1 = {}; v8f acc2 = {}; v8f acc3 = {};

#pragma unroll 4
    for (int k = 0; k < KCHUNK; k += 4) {
        v2f a  = *(const v2f*)(Sp + k);
        v2f b0 = *(const v2f*)(Fp + k);
        v2f b1 = *(const v2f*)(Fp + (size_t)16 * HW + k);
        v2f b2 = *(const v2f*)(Fp + (size_t)32 * HW + k);
        v2f b3 = *(const v2f*)(Fp + (size_t)48 * HW + k);
        acc0 = __builtin_amdgcn_wmma_f32_16x16x4_f32(false, a, false, b0, (short)0, acc0, false, false);
        acc1 = __builtin_amdgcn_wmma_f32_16x16x4_f32(false, a, false, b1, (short)0, acc1, false, false);
        acc2 = __builtin_amdgcn_wmma_f32_16x16x4_f32(false, a, false, b2, (short)0, acc2, false, false);
        acc3 = __builtin_amdgcn_wmma_f32_16x16x4_f32(false, a, false, b3, (short)0, acc3, false, false);
    }

    // Stage per-wave partial tiles in LDS: [wave][vgpr-slot 0..31][lane]
    __shared__ float lds[8 * 1024];
    float* lp = lds + wave * 1024;
#pragma unroll
    for (int j = 0; j < 8; ++j) {
        lp[(0 * 8 + j) * 32 + lane] = acc0[j];
        lp[(1 * 8 + j) * 32 + lane] = acc1[j];
        lp[(2 * 8 + j) * 32 + lane] = acc2[j];
        lp[(3 * 8 + j) * 32 + lane] = acc3[j];
    }
    __syncthreads();

    // Sum the 8 K-chunk partials; 1024 outputs / 256 threads = 4 each.
#pragma unroll
    for (int e = 0; e < 4; ++e) {
        int idx = threadIdx.x * 4 + e;     // flat (slot, lane) index
        float v = 0.0f;
#pragma unroll
        for (int w = 0; w < 8; ++w) v += lds[w * 1024 + idx];
        int slot = idx >> 5;               // t*8 + j
        int l    = idx & 31;
        int t    = slot >> 3;
        int j    = slot & 7;
        int mm   = j + ((l >> 4) << 3);    // C/D layout: VGPR j holds M=j / M=j+8
        int nn   = t * 16 + (l & 15);
        WX[((size_t)b * KPAD + mm) * CIN + c0 + nn] = v;
    }
}

// res = wx - a_sum * centroid ; L2-normalize over the 512 channels ; write out
__global__ __launch_bounds__(256) void vlad_finalize(const float* __restrict__ WX,
                                                     const float* __restrict__ asum,
                                                     const float* __restrict__ centroids,
                                                     float* __restrict__ out) {
    const int b   = blockIdx.x >> 3;
    const int kc  = blockIdx.x & 7;
    const int tid = threadIdx.x;

    const float a = asum[b * KPAD + kc];
    const float* wp = WX + ((size_t)b * KPAD + kc) * CIN;
    const float* cp = centroids + (size_t)kc * CIN;

    float r0 = wp[tid]       - a * cp[tid];
    float r1 = wp[tid + 256] - a * cp[tid + 256];

    __shared__ float red[256];
    red[tid] = r0 * r0 + r1 * r1;
    __syncthreads();
    for (int off = 128; off > 0; off >>= 1) {
        if (tid < off) red[tid] += red[tid + off];
        __syncthreads();
    }
    __shared__ float invn;
    if (tid == 0) invn = 1.0f / fmaxf(sqrtf(red[0]), 1e-12f);
    __syncthreads();

    float* op = out + ((size_t)b * KKEEP + kc) * CIN;
    op[tid]       = r0 * invn;
    op[tid + 256] = r1 * invn;
}

extern "C" void kernel_launch(void* const* d_in, const int* in_sizes, int n_in,
                              void* d_out, int out_size, void* d_ws, size_t ws_size,
                              hipStream_t stream) {
    const float* features  = (const float*)d_in[0];   // (32, 512, 16, 256)
    const float* score     = (const float*)d_in[1];   // (32, 10, 16, 256)
    const float* centroids = (const float*)d_in[2];   // (10, 512)
    float* out = (float*)d_out;                       // (32, 8*512)

    float* S    = (float*)d_ws;                       // 2,097,152 floats
    float* WX   = S  + (size_t)BATCH * KPAD * HW;     //   262,144 floats
    float* ASUM = WX + (size_t)BATCH * KPAD * CIN;    //       512 floats

    vlad_zero_asum<<<2, 256, 0, stream>>>(ASUM);
    vlad_softmax  <<<BATCH * HW / 256, 256, 0, stream>>>(score, S, ASUM);
    vlad_wmma_gemm<<<BATCH * 8,        256, 0, stream>>>(features, S, WX);
    vlad_finalize <<<BATCH * KKEEP,    256, 0, stream>>>(WX, ASUM, centroids, out);
}